// GAT_27960237097248
// MI455X (gfx1250) — compile-verified
//
#include <hip/hip_runtime.h>
#include <math.h>

// ---------------------------------------------------------------------------
// GAT reference on MI455X (gfx1250).
// Dead-code analysis: output = tanh(edges) depends ONLY on the two edge-GAT
// layers (node path never feeds edges). Edge layer reduces (source bugs
// replicated: split==0 -> lp=0, lc = h*sum(a_e)) to 4 batched 256^3 f32 GEMMs
// per layer -> implemented with V_WMMA_F32_16X16X4_F32 (f32 precision kept,
// problem is L2-resident so no payoff from fp16/fp8 conversion).
// ---------------------------------------------------------------------------

#define NN     256
#define BB     4
#define ALPHA  0.2f
#define SLOT   (BB * NN * NN)          // floats per [B,N,N] buffer (1 MiB)

typedef __attribute__((ext_vector_type(2))) float v2f;
typedef __attribute__((ext_vector_type(8))) float v8f;

__device__ __forceinline__ float lrelu(float x) { return x >= 0.f ? x : ALPHA * x; }

// ---- tiny reduction: S = sum(a_e) -----------------------------------------
__global__ void gat_sum_vec(const float* __restrict__ a, int n, float* __restrict__ out) {
    __shared__ float sh[256];
    float s = 0.f;
    for (int i = threadIdx.x; i < n; i += 256) s += a[i];
    sh[threadIdx.x] = s;
    __syncthreads();
    for (int w = 128; w > 0; w >>= 1) {
        if (threadIdx.x < w) sh[threadIdx.x] += sh[threadIdx.x + w];
        __syncthreads();
    }
    if (threadIdx.x == 0) out[0] = sh[0];
}

// ---- C[b] = A[b] @ W + bias  (A:[B,N,N] batched, W:[N,N], bias:[N]) --------
// One wave per 16x16 output tile, K accumulated via v_wmma_f32_16x16x4_f32.
__global__ void gat_gemm_nn_bias(const float* __restrict__ A, const float* __restrict__ W,
                                 const float* __restrict__ bias, float* __restrict__ C) {
    const int lane = threadIdx.x;                 // 32 threads / block
    const int tm = blockIdx.x, tn = blockIdx.y, b = blockIdx.z;
    const float* Ab = A + (size_t)b * NN * NN;
    float*       Cb = C + (size_t)b * NN * NN;

    const int r    = lane & 15;
    const int half = lane >> 4;
    const int kb   = half * 2;
    const int rowA = tm * 16 + r;                 // A row for this lane
    const int colB = tn * 16 + r;                 // B col for this lane

    v8f acc = {};
#pragma unroll 8
    for (int k0 = 0; k0 < NN; k0 += 4) {
        const v2f a = *(const v2f*)(Ab + rowA * NN + k0 + kb);   // b64, contiguous
        v2f bf;
        bf.x = W[(k0 + kb)     * NN + colB];
        bf.y = W[(k0 + kb + 1) * NN + colB];
        acc = __builtin_amdgcn_wmma_f32_16x16x4_f32(false, a, false, bf,
                                                    (short)0, acc, false, false);
    }
    const int cc    = tn * 16 + (lane & 15);
    const int rbase = tm * 16 + half * 8;
    const float bv  = bias ? bias[cc] : 0.f;
#pragma unroll
    for (int i = 0; i < 8; ++i)
        Cb[(size_t)(rbase + i) * NN + cc] = acc[i] + bv;
}

// ---- adj[b,i,j] = (sigmoid(G[i,j]) + sigmoid(G[j,i]) > 1) ------------------
__global__ void gat_make_adj(const float* __restrict__ G, float* __restrict__ ADJ) {
    int idx = blockIdx.x * blockDim.x + threadIdx.x;
    if (idx >= SLOT) return;
    int b = idx / (NN * NN);
    int r = idx - b * NN * NN;
    int i = r / NN, j = r - i * NN;
    const float* Gb = G + (size_t)b * NN * NN;
    float s1 = 1.f / (1.f + __expf(-Gb[i * NN + j]));
    float s2 = 1.f / (1.f + __expf(-Gb[j * NN + i]));
    ADJ[idx] = (s1 + s2 > 1.0f) ? 1.0f : 0.0f;
}

// ---- per-row (b,k): m = max_j lrelu(S*H), rm = mean_j H --------------------
__global__ void gat_row_stats(const float* __restrict__ H, const float* __restrict__ Sp,
                              float* __restrict__ M, float* __restrict__ RM) {
    const int row = blockIdx.x;                   // b*N + k, one wave per row
    const float* h = H + (size_t)row * NN;
    const float S = Sp[0];
    float mx = -3.0e38f, sm = 0.f;
    for (int j = threadIdx.x; j < NN; j += 32) {
        float v = h[j];
        sm += v;
        mx = fmaxf(mx, lrelu(S * v));
    }
#pragma unroll
    for (int off = 16; off > 0; off >>= 1) {
        mx = fmaxf(mx, __shfl_xor(mx, off, 32));
        sm += __shfl_xor(sm, off, 32);
    }
    if (threadIdx.x == 0) { M[row] = mx; RM[row] = sm * (1.0f / NN); }
}

// ---- E = exp(lrelu(S*H) - m[row]);  EH = E*H -------------------------------
__global__ void gat_make_E(const float* __restrict__ H, const float* __restrict__ Sp,
                           const float* __restrict__ M,
                           float* __restrict__ E, float* __restrict__ EH) {
    int idx = blockIdx.x * blockDim.x + threadIdx.x;
    if (idx >= SLOT) return;
    int row = idx / NN;                           // b*N + k
    float v = H[idx];
    float e = expf(lrelu(Sp[0] * v) - M[row]);
    E[idx]  = e;
    EH[idx] = e * v;
}

// ---- NUM[k,i] = sum_j EH[k,j]*adj[i,j];  DEN[k,i] = sum_j E[k,j]*adj[i,j] --
// Fused dual WMMA-NT GEMM: B fragment (adj, contiguous in j) shared by both.
__global__ void gat_gemm_nt_dual(const float* __restrict__ EHm, const float* __restrict__ Em,
                                 const float* __restrict__ ADJ,
                                 float* __restrict__ NUM, float* __restrict__ DEN) {
    const int lane = threadIdx.x;
    const int tm = blockIdx.x, tn = blockIdx.y, b = blockIdx.z;
    const size_t base = (size_t)b * NN * NN;
    const float* EHb = EHm + base;
    const float* Eb  = Em  + base;
    const float* Ab  = ADJ + base;

    const int r    = lane & 15;
    const int half = lane >> 4;
    const int kb   = half * 2;
    const int rowK = tm * 16 + r;                 // A rows (k)
    const int colI = tn * 16 + r;                 // B cols (i) -> adj row i

    v8f accN = {};
    v8f accD = {};
#pragma unroll 8
    for (int j0 = 0; j0 < NN; j0 += 4) {
        const v2f aeh = *(const v2f*)(EHb + rowK * NN + j0 + kb);
        const v2f ae  = *(const v2f*)(Eb  + rowK * NN + j0 + kb);
        const v2f bf  = *(const v2f*)(Ab  + colI * NN + j0 + kb);  // adj[i][j], contig in j
        accN = __builtin_amdgcn_wmma_f32_16x16x4_f32(false, aeh, false, bf,
                                                     (short)0, accN, false, false);
        accD = __builtin_amdgcn_wmma_f32_16x16x4_f32(false, ae,  false, bf,
                                                     (short)0, accD, false, false);
    }
    const int cc    = tn * 16 + (lane & 15);
    const int rbase = tm * 16 + half * 8;
#pragma unroll
    for (int i = 0; i < 8; ++i) {
        NUM[base + (size_t)(rbase + i) * NN + cc] = accN[i];
        DEN[base + (size_t)(rbase + i) * NN + cc] = accD[i];
    }
}

// ---- e'[b,k,i] = 0.5*(Y[k,i] + Y[i,k]),  Y = NUM/DEN or row-mean fallback --
__global__ void gat_combine_sym(const float* __restrict__ NUM, const float* __restrict__ DEN,
                                const float* __restrict__ RM, float* __restrict__ Eout) {
    int idx = blockIdx.x * blockDim.x + threadIdx.x;
    if (idx >= SLOT) return;
    int b = idx / (NN * NN);
    int r = idx - b * NN * NN;
    int k = r / NN, i = r - k * NN;
    size_t base = (size_t)b * NN * NN;
    float n1 = NUM[base + (size_t)k * NN + i], d1 = DEN[base + (size_t)k * NN + i];
    float n2 = NUM[base + (size_t)i * NN + k], d2 = DEN[base + (size_t)i * NN + k];
    float y1 = (d1 > 0.f) ? n1 / d1 : RM[b * NN + k];
    float y2 = (d2 > 0.f) ? n2 / d2 : RM[b * NN + i];
    Eout[idx] = 0.5f * (y1 + y2);
}

// ---- out = tanh(edges), duplicated (reference returns (out, out)) ----------
__global__ void gat_tanh_out(const float* __restrict__ Ein, float* __restrict__ out) {
    int idx = blockIdx.x * blockDim.x + threadIdx.x;
    if (idx >= SLOT) return;
    float t = tanhf(Ein[idx]);
    out[idx]        = t;
    out[idx + SLOT] = t;
}

extern "C" void kernel_launch(void* const* d_in, const int* in_sizes, int n_in,
                              void* d_out, int out_size, void* d_ws, size_t ws_size,
                              hipStream_t stream) {
    (void)in_sizes; (void)n_in; (void)out_size; (void)ws_size;
    const float* edges0 = (const float*)d_in[1];
    float* ws  = (float*)d_ws;
    // Workspace: 7 x 1MiB slots + scalars (~7.4 MB total).
    float* G   = ws + 0 * (size_t)SLOT;   // adj logits, later reused as NUM
    float* ADJ = ws + 1 * (size_t)SLOT;
    float* H   = ws + 2 * (size_t)SLOT;   // e@wp+bp, later reused as DEN
    float* E   = ws + 3 * (size_t)SLOT;
    float* EH  = ws + 4 * (size_t)SLOT;
    float* E1  = ws + 5 * (size_t)SLOT;   // edges after layer 0
    float* E2  = ws + 6 * (size_t)SLOT;   // edges after layer 1
    float* S   = ws + 7 * (size_t)SLOT;   // scalar sum(a_e)
    float* M   = S + 16;                  // [B*N] per-row max
    float* RM  = M + BB * NN;             // [B*N] per-row mean

    const int pbase[2] = {7, 17};         // wp_e, bp_e, a_e, wadj_e, badj_e
    const float* src = edges0;
    float* dst[2] = {E1, E2};

    const dim3 tgrid(NN / 16, NN / 16, BB);   // one wave per 16x16 tile
    const int  eblocks = SLOT / 256;

    for (int l = 0; l < 2; ++l) {
        const float* wp = (const float*)d_in[pbase[l] + 0];
        const float* bp = (const float*)d_in[pbase[l] + 1];
        const float* ae = (const float*)d_in[pbase[l] + 2];
        const float* wa = (const float*)d_in[pbase[l] + 3];
        const float* ba = (const float*)d_in[pbase[l] + 4];

        gat_sum_vec<<<1, 256, 0, stream>>>(ae, 2 * NN, S);
        gat_gemm_nn_bias<<<tgrid, 32, 0, stream>>>(src, wa, ba, G);
        gat_make_adj<<<eblocks, 256, 0, stream>>>(G, ADJ);
        gat_gemm_nn_bias<<<tgrid, 32, 0, stream>>>(src, wp, bp, H);
        gat_row_stats<<<BB * NN, 32, 0, stream>>>(H, S, M, RM);
        gat_make_E<<<eblocks, 256, 0, stream>>>(H, S, M, E, EH);
        gat_gemm_nt_dual<<<tgrid, 32, 0, stream>>>(EH, E, ADJ, /*NUM=*/G, /*DEN=*/H);
        gat_combine_sym<<<eblocks, 256, 0, stream>>>(G, H, RM, dst[l]);
        src = dst[l];
    }
    gat_tanh_out<<<eblocks, 256, 0, stream>>>(E2, (float*)d_out);
}